// GCNNet_61718680043588
// MI455X (gfx1250) — compile-verified
//
#include <hip/hip_runtime.h>

#define NODES  65536
#define FEAT   128
#define EDGES  1048576
#define NGRAPH 64
#define NPGN   1024
#define OUTD   128

typedef float v2f __attribute__((ext_vector_type(2)));
typedef float v8f __attribute__((ext_vector_type(8)));

__device__ __forceinline__ v8f wmma4(v2f a, v2f b, v8f c) {
    // V_WMMA_F32_16X16X4_F32 : D = A(16x4) * B(4x16) + C(16x16), fp32 throughout
    return __builtin_amdgcn_wmma_f32_16x16x4_f32(false, a, false, b, (short)0, c, false, false);
}

// ---------------- degree / norm ----------------
__global__ void k_fill(float* p, float v, int n) {
    int i = blockIdx.x * blockDim.x + threadIdx.x;
    if (i < n) p[i] = v;
}

__global__ void k_degcount(const long long* dst, float* deg, int e) {
    int i = blockIdx.x * blockDim.x + threadIdx.x;
    if (i < e) atomicAdd(&deg[(int)dst[i]], 1.0f);
}

__global__ void k_rsqrt(float* p, int n) {
    int i = blockIdx.x * blockDim.x + threadIdx.x;
    if (i < n) p[i] = rsqrtf(p[i]);
}

// ---------------- weight packing into WMMA B-fragment order --------------------------
// Wp[r][n][2] with r = kstep*2 + half:  Wp[r*2*Nc + 2n + j] = W[4*kstep + 2*half + j][n]
// -> one aligned b64 load per B fragment in the GEMM inner loop.
__global__ void pack_w(const float* __restrict__ W, float* __restrict__ Wp,
                       int Nc, int total) {          // total = (K/2)*Nc
    int i = blockIdx.x * blockDim.x + threadIdx.x;
    if (i >= total) return;
    int n = i % Nc;
    int r = i / Nc;
    int krow = ((r >> 1) << 2) + ((r & 1) << 1);
    float2 v = { W[(size_t)krow * Nc + n], W[(size_t)(krow + 1) * Nc + n] };
    *(float2*)(Wp + (size_t)r * 2 * Nc + 2 * n) = v;
}

// ---------------- generic fp32 WMMA GEMM: C[M,Nc] = A[M,K] @ Wp(packed) ---------------
// one wave -> 32 rows x 64 cols : 8 WMMA per k-step fed by 2 A-b64 + 4 B-b64 loads
__global__ __launch_bounds__(256) void gemm_wmma(const float* __restrict__ A,
                                                 const float* __restrict__ Wp,
                                                 float* __restrict__ C,
                                                 int M, int K, int Nc) {
    int lane  = threadIdx.x & 31;
    int lrow  = lane & 15;
    int half  = lane >> 4;
    int khalf = half << 1;
    int gw    = blockIdx.x * 8 + (threadIdx.x >> 5);
    int groups = Nc >> 6;
    int m0 = (gw / groups) << 5;           // 32-row strip
    int n0 = (gw % groups) << 6;           // 64-col strip
    if (m0 >= M) return;                   // exact grids -> never taken

    const float* arow0 = A + (size_t)(m0 + lrow) * K + khalf;
    const float* arow1 = arow0 + (size_t)16 * K;
    const float* bbase = Wp + (size_t)half * (Nc << 1) + ((n0 + lrow) << 1);

    v8f a00 = {}, a01 = {}, a02 = {}, a03 = {};
    v8f a10 = {}, a11 = {}, a12 = {}, a13 = {};

    for (int k0 = 0; k0 < K; k0 += 4) {
        v2f fa0 = *(const v2f*)(arow0 + k0);
        v2f fa1 = *(const v2f*)(arow1 + k0);
        const float* bp = bbase + (size_t)k0 * Nc;   // (k0/4)*4*Nc
        v2f b0 = *(const v2f*)(bp);
        v2f b1 = *(const v2f*)(bp + 32);
        v2f b2 = *(const v2f*)(bp + 64);
        v2f b3 = *(const v2f*)(bp + 96);
        a00 = wmma4(fa0, b0, a00);  a01 = wmma4(fa0, b1, a01);
        a02 = wmma4(fa0, b2, a02);  a03 = wmma4(fa0, b3, a03);
        a10 = wmma4(fa1, b0, a10);  a11 = wmma4(fa1, b1, a11);
        a12 = wmma4(fa1, b2, a12);  a13 = wmma4(fa1, b3, a13);
    }

    int hi8 = half << 3;
    float* cb0 = C + (size_t)(m0 + hi8) * Nc + n0 + lrow;
    float* cb1 = cb0 + (size_t)16 * Nc;
    #pragma unroll
    for (int v = 0; v < 8; ++v) {
        size_t r = (size_t)v * Nc;
        cb0[r +  0] = a00[v];  cb0[r + 16] = a01[v];
        cb0[r + 32] = a02[v];  cb0[r + 48] = a03[v];
        cb1[r +  0] = a10[v];  cb1[r + 16] = a11[v];
        cb1[r + 32] = a12[v];  cb1[r + 48] = a13[v];
    }
}

// ---------------- aggregation ----------------
// self-loop + bias:  out[n,c] = h[n,c]*dis[n]^2 + b[c]   (C = 1<<cs)
__global__ void agg_pre(const float* __restrict__ h, const float* __restrict__ dis,
                        const float* __restrict__ b, float* __restrict__ o, int cs) {
    size_t i = (size_t)blockIdx.x * blockDim.x + threadIdx.x;
    int n = (int)(i >> cs), c = (int)(i & ((1u << cs) - 1));
    float d = dis[n];
    o[i] = h[i] * d * d + b[c];
}

// one wave per edge: lanes sweep channels in float4s, atomic-add at dst
__global__ __launch_bounds__(256) void agg_edge(const float* __restrict__ h,
                                                const long long* __restrict__ src,
                                                const long long* __restrict__ dst,
                                                const float* __restrict__ dis,
                                                float* __restrict__ o, int C, int e) {
    int gw = blockIdx.x * 8 + (threadIdx.x >> 5);
    if (gw >= e) return;
    int lane = threadIdx.x & 31;
    int s = (int)src[gw];
    int d = (int)dst[gw];
    float w = dis[s] * dis[d];
    const float4* hs = (const float4*)(h + (size_t)s * C);
    float* ob = o + (size_t)d * C;
    int c4n = C >> 2;
    for (int c4 = lane; c4 < c4n; c4 += 32) {
        float4 v = hs[c4];
        float* p = ob + (c4 << 2);
        atomicAdd(p + 0, v.x * w);
        atomicAdd(p + 1, v.y * w);
        atomicAdd(p + 2, v.z * w);
        atomicAdd(p + 3, v.w * w);
    }
}

__global__ void k_relu4(float4* p, size_t n4) {
    size_t i = (size_t)blockIdx.x * blockDim.x + threadIdx.x;
    if (i < n4) {
        float4 v = p[i];
        v.x = fmaxf(v.x, 0.f); v.y = fmaxf(v.y, 0.f);
        v.z = fmaxf(v.z, 0.f); v.w = fmaxf(v.w, 0.f);
        p[i] = v;
    }
}

// ---------------- global max pool over each graph (x3 >= 0 after relu) ----------------
__global__ void pool_max(const float* __restrict__ x3, float* __restrict__ pooled) {
    int g = blockIdx.x, c = threadIdx.x;               // blockDim = 512
    const float* p = x3 + (size_t)g * NPGN * 512 + c;
    float m = 0.0f;
    for (int n = 0; n < NPGN; ++n) m = fmaxf(m, p[(size_t)n * 512]);
    pooled[g * 512 + c] = m;
}

// ---------------- fused MLP head: out = relu(X@fc1_w + fc1_b) @ fc2_w + fc2_b ----------
// one block per 16-row strip; hidden 16x1024 strip staged in LDS; packed weights
#define HSTR 1032
__global__ __launch_bounds__(256) void head_fused(const float* __restrict__ X,
                                                  const float* __restrict__ fc1wp,
                                                  const float* __restrict__ fc1b,
                                                  const float* __restrict__ fc2wp,
                                                  const float* __restrict__ fc2b,
                                                  float* __restrict__ out) {
    __shared__ float t[16 * HSTR];                     // 66 KB of the 320 KB WGP LDS
    int wave  = threadIdx.x >> 5;
    int lane  = threadIdx.x & 31;
    int lrow  = lane & 15;
    int half  = lane >> 4;
    int khalf = half << 1;
    int hi8   = half << 3;
    int m0    = blockIdx.x << 4;

    // ---- phase 1: FC1 (K=512 -> N=1024), wave owns cols [wave*128, wave*128+128)
    // A fragment loaded once per k-step, reused by 8 column tiles (8 accumulators)
    const float* arow  = X + (size_t)(m0 + lrow) * 512 + khalf;
    const float* bbase = fc1wp + (size_t)half * 2048 + (((wave << 7) + lrow) << 1);
    v8f acc[8];
    #pragma unroll
    for (int tt = 0; tt < 8; ++tt) acc[tt] = (v8f){};

    for (int k0 = 0; k0 < 512; k0 += 4) {
        v2f a = *(const v2f*)(arow + k0);
        const float* bp = bbase + (size_t)k0 * 1024;
        #pragma unroll
        for (int tt = 0; tt < 8; ++tt)
            acc[tt] = wmma4(a, *(const v2f*)(bp + (tt << 5)), acc[tt]);
    }
    #pragma unroll
    for (int tt = 0; tt < 8; ++tt) {
        int n0 = (wave << 7) + (tt << 4);
        float bias = fc1b[n0 + lrow];
        #pragma unroll
        for (int v = 0; v < 8; ++v)
            t[(v + hi8) * HSTR + n0 + lrow] = fmaxf(acc[tt][v] + bias, 0.0f);
    }
    __syncthreads();

    // ---- phase 2: FC2 (K=1024 -> N=128), wave owns one 16x16 tile (A from LDS)
    int n0 = wave << 4;
    v8f acc2 = {};
    const float* trow  = t + lrow * HSTR + khalf;
    const float* bbas2 = fc2wp + (size_t)half * 256 + ((n0 + lrow) << 1);
    for (int k0 = 0; k0 < 1024; k0 += 4) {
        v2f a = *(const v2f*)(trow + k0);
        v2f b = *(const v2f*)(bbas2 + (size_t)k0 * 128);
        acc2 = wmma4(a, b, acc2);
    }
    float bias = fc2b[n0 + lrow];
    #pragma unroll
    for (int v = 0; v < 8; ++v)
        out[(size_t)(m0 + hi8 + v) * 128 + n0 + lrow] = acc2[v] + bias;
}

// ---------------- driver ----------------
extern "C" void kernel_launch(void* const* d_in, const int* in_sizes, int n_in,
                              void* d_out, int out_size, void* d_ws, size_t ws_size,
                              hipStream_t stream) {
    const float*     feats = (const float*)d_in[0];
    const long long* ei    = (const long long*)d_in[1];   // int64 [2, E]
    const long long* esrc  = ei;
    const long long* edst  = ei + EDGES;
    const float* W1   = (const float*)d_in[3];
    const float* b1   = (const float*)d_in[4];
    const float* W2   = (const float*)d_in[5];
    const float* b2   = (const float*)d_in[6];
    const float* W3   = (const float*)d_in[7];
    const float* b3   = (const float*)d_in[8];
    const float* fc1w = (const float*)d_in[9];
    const float* fc1b = (const float*)d_in[10];
    const float* fc2w = (const float*)d_in[11];
    const float* fc2b = (const float*)d_in[12];
    float* out = (float*)d_out;

    float* ws     = (float*)d_ws;
    float* dis    = ws;                                   // N
    float* pooled = ws + NODES;                           // 64*512
    float* buf0   = pooled + NGRAPH * 512;                // N*512 (pre-agg h)
    float* buf1   = buf0 + (size_t)NODES * 512;           // N*512 (post-agg x)
    float* w1p    = buf1 + (size_t)NODES * 512;           // 128*128
    float* w2p    = w1p + 128 * 128;                      // 128*256
    float* w3p    = w2p + 128 * 256;                      // 256*512
    float* fc1wp  = w3p + 256 * 512;                      // 512*1024
    float* fc2wp  = fc1wp + 512 * 1024;                   // 1024*128

    // pack all weights into WMMA B-fragment order
    pack_w<<<(64   * 128  + 255) / 256, 256, 0, stream>>>(W1,   w1p,   128,  64 * 128);
    pack_w<<<(64   * 256  + 255) / 256, 256, 0, stream>>>(W2,   w2p,   256,  64 * 256);
    pack_w<<<(128  * 512  + 255) / 256, 256, 0, stream>>>(W3,   w3p,   512,  128 * 512);
    pack_w<<<(256  * 1024 + 255) / 256, 256, 0, stream>>>(fc1w, fc1wp, 1024, 256 * 1024);
    pack_w<<<(512  * 128  + 255) / 256, 256, 0, stream>>>(fc2w, fc2wp, 128,  512 * 128);

    // degree -> D^-1/2 (self-loop counts as 1)
    k_fill    <<<NODES / 256, 256, 0, stream>>>(dis, 1.0f, NODES);
    k_degcount<<<EDGES / 256, 256, 0, stream>>>(edst, dis, EDGES);
    k_rsqrt   <<<NODES / 256, 256, 0, stream>>>(dis, NODES);

    // layer 1: 128 -> 128
    gemm_wmma<<<(NODES / 32) * (128 / 64) / 8, 256, 0, stream>>>(feats, w1p, buf0, NODES, 128, 128);
    agg_pre  <<<NODES * 128 / 256, 256, 0, stream>>>(buf0, dis, b1, buf1, 7);
    agg_edge <<<EDGES / 8, 256, 0, stream>>>(buf0, esrc, edst, dis, buf1, 128, EDGES);
    k_relu4  <<<NODES * 128 / 4 / 256, 256, 0, stream>>>((float4*)buf1, (size_t)NODES * 128 / 4);

    // layer 2: 128 -> 256
    gemm_wmma<<<(NODES / 32) * (256 / 64) / 8, 256, 0, stream>>>(buf1, w2p, buf0, NODES, 128, 256);
    agg_pre  <<<NODES * 256 / 256, 256, 0, stream>>>(buf0, dis, b2, buf1, 8);
    agg_edge <<<EDGES / 8, 256, 0, stream>>>(buf0, esrc, edst, dis, buf1, 256, EDGES);
    k_relu4  <<<NODES * 256 / 4 / 256, 256, 0, stream>>>((float4*)buf1, (size_t)NODES * 256 / 4);

    // layer 3: 256 -> 512
    gemm_wmma<<<(NODES / 32) * (512 / 64) / 8, 256, 0, stream>>>(buf1, w3p, buf0, NODES, 256, 512);
    agg_pre  <<<NODES * 512 / 256, 256, 0, stream>>>(buf0, dis, b3, buf1, 9);
    agg_edge <<<EDGES / 8, 256, 0, stream>>>(buf0, esrc, edst, dis, buf1, 512, EDGES);
    k_relu4  <<<NODES * 512 / 4 / 256, 256, 0, stream>>>((float4*)buf1, (size_t)NODES * 512 / 4);

    // pooling + heads
    pool_max  <<<NGRAPH, 512, 0, stream>>>(buf1, pooled);
    head_fused<<<NODES / 16, 256, 0, stream>>>(buf1, fc1wp, fc1b, fc2wp, fc2b, out);
    head_fused<<<NGRAPH / 16, 256, 0, stream>>>(pooled, fc1wp, fc1b, fc2wp, fc2b,
                                                out + (size_t)NODES * OUTD);
}